// GraphDecoder_28406913695942
// MI455X (gfx1250) — compile-verified
//
#include <hip/hip_runtime.h>
#include <hip/hip_bf16.h>

#define NN      8192
#define LATENT  256
#define HID     64
#define OUTD    256
#define EPS     1e-8f
#define THRESH  0.2f

typedef __attribute__((ext_vector_type(2))) float v2f;
typedef __attribute__((ext_vector_type(8))) float v8f;

__device__ __forceinline__ v8f wmma_f32_k4(v2f a, v2f b, v8f c) {
    // D = A(16x4) * B(4x16) + C(16x16), fp32, wave32
    return __builtin_amdgcn_wmma_f32_16x16x4_f32(
        /*neg_a=*/false, a, /*neg_b=*/false, b,
        /*c_mod=*/(short)0, c, /*reuse_a=*/false, /*reuse_b=*/false);
}

// ---------------------------------------------------------------- flag reset
__global__ void zero_flag_kernel(int* flag) {
    if (threadIdx.x == 0 && blockIdx.x == 0) *flag = 0;
}

// ---------------------------------------------------------------- row norms
// one wave per row; 256 floats -> 8 per lane (two float4), shfl_xor reduce
__global__ __launch_bounds__(256) void norm_kernel(const float* __restrict__ z,
                                                   float* __restrict__ norms) {
    int row  = blockIdx.x * 8 + (threadIdx.x >> 5);
    int lane = threadIdx.x & 31;
    const float* zr = z + (size_t)row * LATENT + lane * 8;
    float4 p0 = *(const float4*)(zr);
    float4 p1 = *(const float4*)(zr + 4);
    float s = p0.x*p0.x + p0.y*p0.y + p0.z*p0.z + p0.w*p0.w
            + p1.x*p1.x + p1.y*p1.y + p1.z*p1.z + p1.w*p1.w;
    #pragma unroll
    for (int off = 16; off > 0; off >>= 1) s += __shfl_xor(s, off, 32);
    if (lane == 0) norms[row] = sqrtf(s);
}

// ---------------------------------------------------------------- decoder
// per wave: 16 rows. GEMM1 (K=256 -> 16x64) via WMMA, bias+relu into padded
// LDS, then GEMM2 (K=64 -> 16x256) via WMMA in 4 groups of 4 N-tiles.
#define HSTR 68   // LDS row stride (floats): (4r+c) mod 64 spreads banks
__global__ __launch_bounds__(128) void decoder_kernel(
    const float* __restrict__ z,  const float* __restrict__ W1,
    const float* __restrict__ b1, const float* __restrict__ W2,
    const float* __restrict__ b2, float* __restrict__ xout) {

    __shared__ float hl[4][16 * HSTR];
    int lane = threadIdx.x & 31;
    int w    = threadIdx.x >> 5;
    int rowBase = blockIdx.x * 64 + w * 16;

    int rsel  = lane & 15;
    int khalf = (lane >> 4) << 1;         // 0 or 2
    int mhalf = (lane >> 4) << 3;         // 0 or 8  (C-matrix M offset)

    // --- GEMM1: h = relu(z * W1 + b1)
    v8f acc[4] = {};
    const float* za = z + (size_t)(rowBase + rsel) * LATENT;
    for (int k = 0; k < LATENT; k += 4) {
        int kk = k + khalf;
        v2f a = *(const v2f*)(za + kk);
        #pragma unroll
        for (int nt = 0; nt < 4; ++nt) {
            int n = nt * 16 + rsel;
            v2f b;
            b.x = W1[kk * HID + n];
            b.y = W1[(kk + 1) * HID + n];
            acc[nt] = wmma_f32_k4(a, b, acc[nt]);
        }
    }
    float* hw = &hl[w][0];
    #pragma unroll
    for (int nt = 0; nt < 4; ++nt) {
        int n = nt * 16 + rsel;
        float bias = b1[n];
        #pragma unroll
        for (int r = 0; r < 8; ++r) {
            float v = acc[nt][r] + bias;
            hw[(mhalf + r) * HSTR + n] = v > 0.f ? v : 0.f;
        }
    }
    __syncthreads();

    // --- GEMM2: x = h * W2 + b2   (4 groups of 4 N-tiles)
    for (int g = 0; g < 4; ++g) {
        v8f acc2[4] = {};
        for (int k = 0; k < HID; k += 4) {
            int kk = k + khalf;
            v2f a = *(const v2f*)(hw + rsel * HSTR + kk);
            #pragma unroll
            for (int nt = 0; nt < 4; ++nt) {
                int n = (g * 4 + nt) * 16 + rsel;
                v2f b;
                b.x = W2[kk * OUTD + n];
                b.y = W2[(kk + 1) * OUTD + n];
                acc2[nt] = wmma_f32_k4(a, b, acc2[nt]);
            }
        }
        #pragma unroll
        for (int nt = 0; nt < 4; ++nt) {
            int n = (g * 4 + nt) * 16 + rsel;
            float bias = b2[n];
            #pragma unroll
            for (int r = 0; r < 8; ++r)
                xout[(size_t)(rowBase + mhalf + r) * OUTD + n] = acc2[nt][r] + bias;
        }
    }
}

// ---------------------------------------------------------------- sim GEMM
// 128x128 output tile per block (256 thr / 8 waves, 4x2 wave grid, each wave
// 32x64 = 2x4 WMMA accumulators). z panels staged in LDS, K chunked by 32.
#define BK   32
#define LSTR 36   // padded LDS stride (floats): conflict-free b64 frag loads
__global__ __launch_bounds__(256) void sim_kernel(
    const float* __restrict__ z, const float* __restrict__ norms,
    float* __restrict__ sim, float* __restrict__ mask, int* __restrict__ flag) {

    __shared__ float As[128 * LSTR];
    __shared__ float Bs[128 * LSTR];

    int tid  = threadIdx.x;
    int lane = tid & 31;
    int w    = tid >> 5;
    int wm   = w & 3;           // 4 M-strips of 32
    int wn   = w >> 2;          // 2 N-strips of 64
    int rowBlock = blockIdx.x * 128;
    int colBlock = blockIdx.y * 128;

    int rsel  = lane & 15;
    int khalf = (lane >> 4) << 1;
    int mhalf = (lane >> 4) << 3;

    v8f acc[2][4] = {};

    for (int kc = 0; kc < LATENT; kc += BK) {
        // cooperative fill: 128 rows x 32 floats per panel, float4 per thread
        #pragma unroll
        for (int it = 0; it < 4; ++it) {
            int idx = tid + it * 256;        // 0..1023
            int r   = idx >> 3;              // 0..127
            int c4  = (idx & 7) << 2;        // 0..28
            *(float4*)(As + r * LSTR + c4) =
                *(const float4*)(z + (size_t)(rowBlock + r) * LATENT + kc + c4);
            *(float4*)(Bs + r * LSTR + c4) =
                *(const float4*)(z + (size_t)(colBlock + r) * LATENT + kc + c4);
        }
        __syncthreads();
        #pragma unroll
        for (int k = 0; k < BK; k += 4) {
            int kk = k + khalf;
            v2f a0 = *(const v2f*)(As + (wm * 32 +      rsel) * LSTR + kk);
            v2f a1 = *(const v2f*)(As + (wm * 32 + 16 + rsel) * LSTR + kk);
            v2f b[4];
            #pragma unroll
            for (int nt = 0; nt < 4; ++nt)
                b[nt] = *(const v2f*)(Bs + (wn * 64 + nt * 16 + rsel) * LSTR + kk);
            #pragma unroll
            for (int nt = 0; nt < 4; ++nt) {
                acc[0][nt] = wmma_f32_k4(a0, b[nt], acc[0][nt]);
                acc[1][nt] = wmma_f32_k4(a1, b[nt], acc[1][nt]);
            }
        }
        __syncthreads();
    }

    // epilogue: sim = dot / max(ni*nj, eps); mask = (sim>T) && i!=j
    int anyEdge = 0;
    #pragma unroll
    for (int mt = 0; mt < 2; ++mt) {
        int ibase = rowBlock + wm * 32 + mt * 16 + mhalf;
        float ni[8];
        #pragma unroll
        for (int r = 0; r < 8; ++r) ni[r] = norms[ibase + r];
        #pragma unroll
        for (int nt = 0; nt < 4; ++nt) {
            int j = colBlock + wn * 64 + nt * 16 + rsel;
            float nj = norms[j];
            #pragma unroll
            for (int r = 0; r < 8; ++r) {
                int i = ibase + r;
                float s = acc[mt][nt][r] / fmaxf(ni[r] * nj, EPS);
                size_t off = (size_t)i * NN + j;
                sim[off] = s;
                int e = (s > THRESH) & (i != j);
                mask[off] = e ? 1.0f : 0.0f;
                anyEdge |= e;
            }
        }
    }
    if (anyEdge) atomicOr(flag, 1);
}

// ---------------------------------------------------------------- top-2/row
__global__ __launch_bounds__(256) void top2_kernel(const float* __restrict__ sim,
                                                   int* __restrict__ top2) {
    int row  = blockIdx.x * 8 + (threadIdx.x >> 5);
    int lane = threadIdx.x & 31;
    const float NEG = -3.402823466e38f;
    float v0 = NEG, v1 = NEG; int i0 = 0, i1 = 0;
    const float* sr = sim + (size_t)row * NN;
    for (int j = lane; j < NN; j += 32) {
        float s = (j == row) ? NEG : sr[j];
        if (s > v0)      { v1 = v0; i1 = i0; v0 = s; i0 = j; }
        else if (s > v1) { v1 = s;  i1 = j; }
    }
    #pragma unroll
    for (int off = 16; off > 0; off >>= 1) {
        float ov0 = __shfl_xor(v0, off, 32); int oi0 = __shfl_xor(i0, off, 32);
        float ov1 = __shfl_xor(v1, off, 32); int oi1 = __shfl_xor(i1, off, 32);
        if (ov0 > v0) {
            float t0 = v0; int ti0 = i0;
            v0 = ov0; i0 = oi0;
            if (t0 > ov1) { v1 = t0;  i1 = ti0; }
            else          { v1 = ov1; i1 = oi1; }
        } else if (ov0 > v1) { v1 = ov0; i1 = oi0; }
    }
    if (lane == 0) { top2[2 * row] = i0; top2[2 * row + 1] = i1; }
}

// ---------------------------------------------------------------- fallback
// only taken when no threshold edge exists (flag==0) — early-exits otherwise
__global__ __launch_bounds__(256) void fallback_kernel(const int* __restrict__ flag,
                                                       const int* __restrict__ top2,
                                                       float* __restrict__ mask) {
    if (*flag) return;
    int i = blockIdx.x;
    int t0 = top2[2 * i], t1 = top2[2 * i + 1];
    for (int j = threadIdx.x; j < NN; j += blockDim.x) {
        int u0 = top2[2 * j], u1 = top2[2 * j + 1];
        int e = (j == t0) | (j == t1) | (i == u0) | (i == u1);
        mask[(size_t)i * NN + j] = e ? 1.0f : 0.0f;
    }
}

// ---------------------------------------------------------------- launch
extern "C" void kernel_launch(void* const* d_in, const int* in_sizes, int n_in,
                              void* d_out, int out_size, void* d_ws, size_t ws_size,
                              hipStream_t stream) {
    const float* z  = (const float*)d_in[0];
    const float* W1 = (const float*)d_in[1];
    const float* b1 = (const float*)d_in[2];
    const float* W2 = (const float*)d_in[3];
    const float* b2 = (const float*)d_in[4];

    float* x_recon = (float*)d_out;                       // [N, OUT]
    float* sim     = x_recon + (size_t)NN * OUTD;         // [N, N]
    float* mask    = sim + (size_t)NN * NN;               // [N, N] (0/1 floats)

    int*   flag  = (int*)d_ws;
    float* norms = (float*)((char*)d_ws + 256);
    int*   top2  = (int*)((char*)d_ws + 256 + NN * sizeof(float));

    zero_flag_kernel<<<1, 1, 0, stream>>>(flag);
    norm_kernel<<<NN / 8, 256, 0, stream>>>(z, norms);
    decoder_kernel<<<NN / 64, 128, 0, stream>>>(z, W1, b1, W2, b2, x_recon);

    dim3 g(NN / 128, NN / 128);
    sim_kernel<<<g, 256, 0, stream>>>(z, norms, sim, mask, flag);

    top2_kernel<<<NN / 8, 256, 0, stream>>>(sim, top2);
    fallback_kernel<<<NN, 256, 0, stream>>>(flag, top2, mask);
}